// MyLayer_18073222382197
// MI455X (gfx1250) — compile-verified
//
#include <hip/hip_runtime.h>
#include <stdint.h>

// FFM pairwise interaction:
//   W[i,j]  = sum_k kernel[i, fid[j], k] * kernel[j, fid[i], k]   (strict upper)
//   out[b]  = sum_{i<j} x[b,i] * W[i,j] * x[b,j]
// Implemented as GEMM Y = X * Wm (fp32 WMMA 16x16x4) fused with row-dot.

typedef float v2f __attribute__((ext_vector_type(2)));
typedef float v8f __attribute__((ext_vector_type(8)));

#define BATCH     32768
#define D_IN      500
#define DPAD      512
#define NFIELD    40
#define NOUT      30
#define BSTRIDE   516            // padded LDS row stride in floats (bank-conflict free)
#define ROWS_PER_BLOCK 128       // 8 waves * 16 rows

// ---------------------------------------------------------------------------
// Kernel 1: build transposed masked interaction matrix
//   WT[n*DPAD + k] = (k<n, both <500) ? dot30(kernel[k, fid[n]], kernel[n, fid[k]]) : 0
// ---------------------------------------------------------------------------
__global__ __launch_bounds__(256) void build_wt(const float* __restrict__ kern,
                                                const int*   __restrict__ fid,
                                                float*       __restrict__ wt) {
  int idx = blockIdx.x * 256 + threadIdx.x;   // 0 .. 512*512-1
  int k = idx & (DPAD - 1);                   // row of W   (K index of GEMM)
  int n = idx >> 9;                           // col of W   (N index of GEMM)
  float v = 0.0f;
  if (k < D_IN && n < D_IN && k < n) {
    const float* a = kern + (size_t)k * (NFIELD * NOUT) + fid[n] * NOUT;
    const float* b = kern + (size_t)n * (NFIELD * NOUT) + fid[k] * NOUT;
#pragma unroll
    for (int t = 0; t < NOUT; ++t) v += a[t] * b[t];
  }
  wt[(size_t)n * DPAD + k] = v;
}

// ---------------------------------------------------------------------------
// Kernel 2: fused GEMM + row-dot using V_WMMA_F32_16X16X4_F32.
//   Each wave computes a 16-row stripe of Y = X*Wm across all 512 columns,
//   multiplying each 16x16 Y tile by x[b, n] on the fly. WT slabs are staged
//   into LDS with CDNA5 async global->LDS copies (ASYNCcnt).
// ---------------------------------------------------------------------------
__global__ __launch_bounds__(256) void ffm_quadform(const float* __restrict__ X,
                                                    const float* __restrict__ WT,
                                                    float*       __restrict__ out) {
  __shared__ float btile[16 * BSTRIDE];       // 33 KB: one 16-column slab of WT

  const int lane = threadIdx.x & 31;
  const int wave = threadIdx.x >> 5;
  const int half = lane >> 4;                 // 0: lanes 0-15, 1: lanes 16-31
  const int lm   = lane & 15;
  const int m0   = blockIdx.x * ROWS_PER_BLOCK + wave * 16;

  // A fragment source: lane holds A[M=lm][K = k + 2*half + {0,1}]
  const float* arow = X + (size_t)(m0 + lm) * D_IN + 2 * half;

  // Per-thread async-copy addresses for the 16x512 slab (8 x B128 each).
  // Slab float4 index f = tid + r*256; row = f>>7, c4 = f&127.
  const int srow = threadIdx.x >> 7;          // base row (0 or 1); +2 per r
  const int sc4  = threadIdx.x & 127;
  const unsigned gbase = (unsigned)(srow * (DPAD * 4) + sc4 * 16);          // bytes
  const unsigned lbase = (unsigned)(uintptr_t)btile                         // LDS addr
                       + (unsigned)(srow * (BSTRIDE * 4) + sc4 * 16);

  float po[8];
#pragma unroll
  for (int v = 0; v < 8; ++v) po[v] = 0.0f;

  for (int n0 = 0; n0 < DPAD; n0 += 16) {
    __syncthreads();   // previous tile's LDS reads complete

    // Async stage: 16 rows x 512 floats, direct global->LDS, no VGPR traffic.
    {
      const float* src = WT + (size_t)n0 * DPAD;   // uniform -> SGPR pair
#pragma unroll
      for (int r = 0; r < 8; ++r) {
        unsigned gl = gbase + (unsigned)(r * 2 * DPAD * 4);     // +2 rows per r
        unsigned ll = lbase + (unsigned)(r * 2 * BSTRIDE * 4);
        asm volatile("global_load_async_to_lds_b128 %0, %1, %2"
                     :: "v"(ll), "v"(gl), "s"(src) : "memory");
      }
      asm volatile("s_wait_asynccnt 0" ::: "memory");
    }
    // Prefetch next slab toward L2 while we compute on this one.
    if (n0 + 16 < DPAD)
      __builtin_prefetch(WT + (size_t)(n0 + 16) * DPAD, 0, 0);

    __syncthreads();

    // B fragment source in LDS: lane holds B[K = k + 2*half + {0,1}][N=lm]
    const float* brow = btile + lm * BSTRIDE + 2 * half;

    v8f acc = {};
#pragma unroll 5
    for (int k = 0; k < D_IN; k += 4) {       // 500 = 125 * 4, exact
      v2f a = *(const v2f*)(arow + k);
      v2f b = *(const v2f*)(brow + k);
      acc = __builtin_amdgcn_wmma_f32_16x16x4_f32(
          /*neg_a=*/false, a, /*neg_b=*/false, b,
          /*c_mod=*/(short)0, acc, /*reuse_a=*/false, /*reuse_b=*/false);
    }

    // Fused row-dot: acc[v] (lane) = Y[m0 + v + 8*half][n0 + lm].
    // Branch-free: clamp column and apply a multiplicative mask.
    const int   col   = n0 + lm;
    const int   colc  = (col < D_IN) ? col : (D_IN - 1);
    const float cmask = (col < D_IN) ? 1.0f : 0.0f;
    const float* xcol = X + (size_t)(m0 + 8 * half) * D_IN + colc;
#pragma unroll
    for (int v = 0; v < 8; ++v) {
      float xv = xcol[(size_t)v * D_IN];
      po[v] += acc[v] * (xv * cmask);
    }
  }

  // Reduce across the 16 lanes of each half-wave (xor masks stay in-group).
#pragma unroll
  for (int v = 0; v < 8; ++v) {
    float s = po[v];
    s += __shfl_xor(s, 1, 32);
    s += __shfl_xor(s, 2, 32);
    s += __shfl_xor(s, 4, 32);
    s += __shfl_xor(s, 8, 32);
    po[v] = s;
  }
  if (lm == 0) {
#pragma unroll
    for (int v = 0; v < 8; ++v) out[m0 + 8 * half + v] = po[v];
  }
}

// ---------------------------------------------------------------------------
extern "C" void kernel_launch(void* const* d_in, const int* in_sizes, int n_in,
                              void* d_out, int out_size, void* d_ws, size_t ws_size,
                              hipStream_t stream) {
  const float* x    = (const float*)d_in[0];   // (32768, 500) f32
  const float* kern = (const float*)d_in[1];   // (500, 40, 30) f32
  const int*   fid  = (const int*)d_in[2];     // (500,) int
  float* out = (float*)d_out;                  // (32768,) f32
  float* wt  = (float*)d_ws;                   // 512*512 f32 scratch (1 MB)

  build_wt<<<(DPAD * DPAD) / 256, 256, 0, stream>>>(kern, fid, wt);
  ffm_quadform<<<BATCH / ROWS_PER_BLOCK, 256, 0, stream>>>(x, wt, out);
}